// UpsampleLoss_15908558864626
// MI455X (gfx1250) — compile-verified
//
#include <hip/hip_runtime.h>
#include <math.h>

// UpsampleLoss for MI455X (gfx1250, wave32).
//
// Pairwise sq-dists fully inside V_WMMA_F32_16X16X4_F32:
//   A(16x4) = [-2x_i, -2y_i, -2z_i, 1.0]   (row i)
//   B(4x16) = [ x_j ,  y_j ,  z_j , bb_j]  (col j, bb = |p_j|^2)
//   C       = aa_i broadcast               (aa = |p_i|^2)
//   D = A.B + C = -2 a.b + bb + aa = ||a-b||^2   -- zero epilogue adds.
//
// LDS holds float4 {x,y,z,|p|^2}, so the B operand for every lane is exactly
// the raw float2 at sPts[j] + (hf ? 8B : 0B): lanes 0-15 get {x,y} (K0,K1),
// lanes 16-31 get {z,bb} (K2,K3).  One uniform ds_load_b64 per tile, no VALU.
//
// VGPR layouts (CDNA5 ISA 7.12.2, wave32):
//   A (16x4, 32-bit):  lanes 0-15: M=lane, VGPR0=K0, VGPR1=K1
//                      lanes 16-31: M=lane-16, VGPR0=K2, VGPR1=K3
//   B (4x16, 32-bit):  lanes 0-15: N=lane, VGPR0=K0, VGPR1=K1
//                      lanes 16-31: N=lane-16, VGPR0=K2, VGPR1=K3
//   C/D (16x16 f32):   lanes 0-15: (M=v, N=lane); lanes 16-31: (M=v+8, N=lane-16)

typedef __attribute__((ext_vector_type(2))) float v2f;
typedef __attribute__((ext_vector_type(8))) float v8f;

#define TILE 16
#define MTILES 2                        /* A row-tiles per wave per B build */
#define WAVES_PER_BLOCK 8
#define ROWS_PER_BLOCK (WAVES_PER_BLOCK * MTILES * TILE) /* 256 */
#define MAX_N 4096

#define C_RADIUS 0.07f
#define C_H 0.03f
#define C_EPS 1e-12f
#define C_ALPHA 1.0f
#define FLT_BIG 3.402823466e38f

__device__ __forceinline__ v8f wmma_dot(v2f a, v2f b, v8f c) {
#if __has_builtin(__builtin_amdgcn_wmma_f32_16x16x4_f32)
  // (neg_a, A, neg_b, B, c_mod, C, reuse_a, reuse_b)
  return __builtin_amdgcn_wmma_f32_16x16x4_f32(false, a, false, b, (short)0, c,
                                               false, false);
#else
  // Shuffle-based emulation (compiles even if the builtin is absent).
  int lane = threadIdx.x & 31;
  int lr = lane & 15;
  int hf = lane >> 4;
  float B0 = __shfl(b.x, lr, 32), B1 = __shfl(b.y, lr, 32);
  float B2 = __shfl(b.x, lr + 16, 32), B3 = __shfl(b.y, lr + 16, 32);
#pragma unroll
  for (int v = 0; v < 8; ++v) {
    int M = v + 8 * hf;
    float A0 = __shfl(a.x, M, 32), A1 = __shfl(a.y, M, 32);
    float A2 = __shfl(a.x, M + 16, 32), A3 = __shfl(a.y, M + 16, 32);
    c[v] += A0 * B0 + A1 * B1 + A2 * B2 + A3 * B3;
  }
  return c;
#endif
}

// ---------------------------------------------------------------------------
// Kernel 1: per pred row i, min_j ||pred_i - gt_j||^2; block-partial sums.
// ---------------------------------------------------------------------------
__global__ __launch_bounds__(256) void emd_min_kernel(
    const float* __restrict__ pred, const float* __restrict__ gt,
    float* __restrict__ part, int N, int chunksPerBatch) {
  __shared__ float4 sPts[MAX_N]; // {x, y, z, |p|^2}  == 64 KB

  const int b = blockIdx.x / chunksPerBatch;
  const int chunk = blockIdx.x % chunksPerBatch;

  { // stage gt[b] into LDS with precomputed squared norm
    const float* src = gt + (size_t)b * N * 3;
    for (int p = threadIdx.x; p < N; p += blockDim.x) {
      float x = src[3 * p + 0], y = src[3 * p + 1], z = src[3 * p + 2];
      sPts[p] = make_float4(x, y, z, x * x + y * y + z * z);
    }
  }
  __syncthreads();

  const int wid = threadIdx.x >> 5;
  const int lane = threadIdx.x & 31;
  const int lr = lane & 15;
  const int hf = lane >> 4;
  const int rowBase = chunk * ROWS_PER_BLOCK + wid * (MTILES * TILE);

  v2f Am[MTILES];
  v8f cin[MTILES]; // aa broadcast: cin[m][v] = |pred_{rowBase+m*16+v+8*hf}|^2
  float mn[MTILES][8];
#pragma unroll
  for (int m = 0; m < MTILES; ++m) {
    int i = rowBase + m * TILE + lr;
    int ic = (i < N) ? i : (N - 1);
    const float* P = pred + ((size_t)b * N + ic) * 3;
    float x = P[0], y = P[1], z = P[2];
    float aa = x * x + y * y + z * z;
    float ax = hf ? (-2.0f * z) : (-2.0f * x);
    float ay = hf ? 1.0f : (-2.0f * y); // K3 = 1.0 multiplies B's bb row
    Am[m] = (v2f){ax, ay};
#pragma unroll
    for (int v = 0; v < 8; ++v) {
      cin[m][v] = __shfl(aa, hf ? (v + 8) : v, 32);
      mn[m][v] = FLT_BIG;
    }
  }

  const float* sp = (const float*)sPts;
  const int sel = hf ? 2 : 0;
  for (int j0 = 0; j0 < N; j0 += TILE) {
    // lanes 0-15: {x,y}; lanes 16-31: {z,bb} -- B operand verbatim from LDS
    float2 pq = *reinterpret_cast<const float2*>(sp + 4 * (j0 + lr) + sel);
    v2f Bv = {pq.x, pq.y};
#pragma unroll
    for (int m = 0; m < MTILES; ++m) {
      v8f c = wmma_dot(Am[m], Bv, cin[m]); // c[v] = ||a_row - b_col||^2
#pragma unroll
      for (int v = 0; v < 8; ++v) mn[m][v] = fminf(mn[m][v], c[v]);
    }
  }

  float s = 0.f;
#pragma unroll
  for (int m = 0; m < MTILES; ++m) {
    // min across the 16 lanes of each half (halves hold different M rows)
#pragma unroll
    for (int off = 1; off < 16; off <<= 1)
#pragma unroll
      for (int v = 0; v < 8; ++v)
        mn[m][v] = fminf(mn[m][v], __shfl_xor(mn[m][v], off, 32));
#pragma unroll
    for (int v = 0; v < 8; ++v) {
      int row = rowBase + m * TILE + v + 8 * hf;
      s += (row < N) ? mn[m][v] : 0.f;
    }
  }

  __syncthreads(); // everyone done reading sPts; alias it as reduction scratch
  float* sRed = (float*)sPts;
  if (lr == 0) sRed[wid * 2 + hf] = s;
  __syncthreads();
  if (threadIdx.x == 0) {
    float t = 0.f;
#pragma unroll
    for (int k = 0; k < WAVES_PER_BLOCK * 2; ++k) t += sRed[k];
    part[blockIdx.x] = t;
  }
}

// ---------------------------------------------------------------------------
// Kernel 2: 4 nearest non-self neighbors in pred; repulsion partial sums.
// ---------------------------------------------------------------------------
#define PUSH4(idx, dval)                                          \
  do {                                                            \
    float _x = (dval), _n;                                        \
    _n = fminf(t0[idx], _x); _x = fmaxf(t0[idx], _x); t0[idx] = _n; \
    _n = fminf(t1[idx], _x); _x = fmaxf(t1[idx], _x); t1[idx] = _n; \
    _n = fminf(t2[idx], _x); _x = fmaxf(t2[idx], _x); t2[idx] = _n; \
    t3[idx] = fminf(t3[idx], _x);                                 \
  } while (0)

__global__ __launch_bounds__(256) void rep_knn_kernel(
    const float* __restrict__ pred, float* __restrict__ part, int N,
    int chunksPerBatch) {
  __shared__ float4 sPts[MAX_N]; // {x, y, z, |p|^2}

  const int b = blockIdx.x / chunksPerBatch;
  const int chunk = blockIdx.x % chunksPerBatch;

  { // stage pred[b] into LDS with precomputed squared norm
    const float* src = pred + (size_t)b * N * 3;
    for (int p = threadIdx.x; p < N; p += blockDim.x) {
      float x = src[3 * p + 0], y = src[3 * p + 1], z = src[3 * p + 2];
      sPts[p] = make_float4(x, y, z, x * x + y * y + z * z);
    }
  }
  __syncthreads();

  const int wid = threadIdx.x >> 5;
  const int lane = threadIdx.x & 31;
  const int lr = lane & 15;
  const int hf = lane >> 4;
  const int rowBase = chunk * ROWS_PER_BLOCK + wid * (MTILES * TILE);

  v2f Am[MTILES];
  v8f cin[MTILES];
  bool isSelf[8]; // slot v holds column lr of row (v + 8*hf) within a tile
#pragma unroll
  for (int v = 0; v < 8; ++v) isSelf[v] = ((v + 8 * hf) == lr);

#pragma unroll
  for (int m = 0; m < MTILES; ++m) {
    int i = rowBase + m * TILE + lr;
    int ic = (i < N) ? i : (N - 1);
    float4 pp = sPts[ic];
    float ax = hf ? (-2.0f * pp.z) : (-2.0f * pp.x);
    float ay = hf ? 1.0f : (-2.0f * pp.y);
    Am[m] = (v2f){ax, ay};
#pragma unroll
    for (int v = 0; v < 8; ++v)
      cin[m][v] = __shfl(pp.w, hf ? (v + 8) : v, 32);
  }

  float t0[MTILES * 8], t1[MTILES * 8], t2[MTILES * 8], t3[MTILES * 8];
#pragma unroll
  for (int k = 0; k < MTILES * 8; ++k) t0[k] = t1[k] = t2[k] = t3[k] = FLT_BIG;

  const float* sp = (const float*)sPts;
  const int sel = hf ? 2 : 0;
  for (int j0 = 0; j0 < N; j0 += TILE) {
    float2 pq = *reinterpret_cast<const float2*>(sp + 4 * (j0 + lr) + sel);
    v2f Bv = {pq.x, pq.y};
#pragma unroll
    for (int m = 0; m < MTILES; ++m) {
      const bool diag = (j0 == rowBase + m * TILE); // wave-uniform
      v8f c = wmma_dot(Am[m], Bv, cin[m]);          // c[v] = d2 directly
#pragma unroll
      for (int v = 0; v < 8; ++v) {
        float d2 = (diag && isSelf[v]) ? FLT_BIG : c[v]; // mask self distance
        PUSH4(m * 8 + v, d2);
      }
    }
  }

  // butterfly merge of sorted-4 lists across the 16 lanes of each half
#pragma unroll
  for (int off = 1; off < 16; off <<= 1) {
#pragma unroll
    for (int k = 0; k < MTILES * 8; ++k) {
      float r0 = __shfl_xor(t0[k], off, 32);
      float r1 = __shfl_xor(t1[k], off, 32);
      float r2 = __shfl_xor(t2[k], off, 32);
      float r3 = __shfl_xor(t3[k], off, 32);
      PUSH4(k, r0);
      PUSH4(k, r1);
      PUSH4(k, r2);
      PUSH4(k, r3);
    }
  }

  float s = 0.f;
  const float INV_H2 = 1.0f / (C_H * C_H);
#pragma unroll
  for (int m = 0; m < MTILES; ++m) {
#pragma unroll
    for (int v = 0; v < 8; ++v) {
      int row = rowBase + m * TILE + v + 8 * hf;
      float acc = 0.f;
      float ds[4] = {t0[m * 8 + v], t1[m * 8 + v], t2[m * 8 + v], t3[m * 8 + v]};
#pragma unroll
      for (int k = 0; k < 4; ++k) {
        float d2c = fmaxf(ds[k], C_EPS);
        float dist = sqrtf(d2c);
        float w = __expf(-d2c * INV_H2);
        acc += (C_RADIUS - dist) * w;
      }
      s += (row < N) ? acc : 0.f;
    }
  }

  __syncthreads(); // done reading sPts; alias it as reduction scratch
  float* sRed = (float*)sPts;
  if (lr == 0) sRed[wid * 2 + hf] = s;
  __syncthreads();
  if (threadIdx.x == 0) {
    float t = 0.f;
#pragma unroll
    for (int k = 0; k < WAVES_PER_BLOCK * 2; ++k) t += sRed[k];
    part[blockIdx.x] = t;
  }
}

// ---------------------------------------------------------------------------
// Finalize: emd = 100/B * sum_b acc_b/(3N r_b); rep = alpha * sum/(B*N*4)
// ---------------------------------------------------------------------------
__global__ void finalize_kernel(const float* __restrict__ emdPart,
                                const float* __restrict__ repPart,
                                const float* __restrict__ radius, int B, int N,
                                int chunksPerBatch, float* __restrict__ out) {
  if (threadIdx.x != 0 || blockIdx.x != 0) return;
  float emd = 0.f;
  for (int b = 0; b < B; ++b) {
    float s = 0.f;
    for (int c = 0; c < chunksPerBatch; ++c) s += emdPart[b * chunksPerBatch + c];
    emd += s / (3.0f * (float)N * radius[b]);
  }
  emd = emd / (float)B * 100.0f;
  float rs = 0.f;
  const int G = B * chunksPerBatch;
  for (int k = 0; k < G; ++k) rs += repPart[k];
  float rep = rs / ((float)B * (float)N * 4.0f);
  out[0] = emd;
  out[1] = C_ALPHA * rep;
}

extern "C" void kernel_launch(void* const* d_in, const int* in_sizes, int n_in,
                              void* d_out, int out_size, void* d_ws,
                              size_t ws_size, hipStream_t stream) {
  const float* pred = (const float*)d_in[0];
  const float* gt = (const float*)d_in[1];
  const float* radius = (const float*)d_in[2];
  const int B = in_sizes[2];           // pcd_radius: (B,1)
  const int N = in_sizes[0] / (3 * B); // pred: (B,N,3)
  const int chunks = (N + ROWS_PER_BLOCK - 1) / ROWS_PER_BLOCK;
  const int G = B * chunks;

  float* emdPart = (float*)d_ws; // G floats
  float* repPart = emdPart + G;  // G floats
  (void)n_in; (void)out_size; (void)ws_size;

  emd_min_kernel<<<G, 256, 0, stream>>>(pred, gt, emdPart, N, chunks);
  rep_knn_kernel<<<G, 256, 0, stream>>>(pred, repPart, N, chunks);
  finalize_kernel<<<1, 1, 0, stream>>>(emdPart, repPart, radius, B, N, chunks,
                                       (float*)d_out);
}